// Encoder_47811575939688
// MI455X (gfx1250) — compile-verified
//
#include <hip/hip_runtime.h>

#define N_NODES 50000
#define N_EDGES 1600000
#define IN_DIM  512
#define OUT_DIM 256

typedef float v2f __attribute__((ext_vector_type(2)));
typedef float v8f __attribute__((ext_vector_type(8)));

// ---------------------------------------------------------------------------
// Row-pointer build: lower_bound of each node id in the sorted adj_rows array.
// ---------------------------------------------------------------------------
__global__ void __launch_bounds__(256)
gcn_build_rowptr(const int* __restrict__ rows, int* __restrict__ rowptr,
                 int n_nodes, int n_edges) {
    int i = blockIdx.x * blockDim.x + threadIdx.x;
    if (i > n_nodes) return;
    if (i == n_nodes) { rowptr[i] = n_edges; return; }
    int lo = 0, hi = n_edges;
    while (lo < hi) {
        int mid = (lo + hi) >> 1;
        if (rows[mid] < i) lo = mid + 1; else hi = mid;
    }
    rowptr[i] = lo;
}

// ---------------------------------------------------------------------------
// Dense GEMM: C[M,256] = A[M,K] @ B[K,256] using V_WMMA_F32_16X16X4_F32.
// One wave owns a 16-row M-tile; loops over 4 n-groups of 64 columns,
// holding 4 accumulator tiles (32 VGPRs) and reusing the A fragment.
//
// f32 WMMA fragment layouts (wave32, per ISA 7.12.2):
//   A 16x4 : lane l -> m = l&15 ; VGPR0 = A[m][k0+2*(l>>4)], VGPR1 = +1
//   B 4x16 : lane l -> n = l&15 ; VGPR0 = B[k0+2*(l>>4)][n], VGPR1 = +1
//   C 16x16: lane l -> n = l&15 ; VGPR v = C[v + 8*(l>>4)][n]
// ---------------------------------------------------------------------------
__global__ void __launch_bounds__(128)
gcn_gemm_wmma_f32(const float* __restrict__ A, const float* __restrict__ B,
                  float* __restrict__ C, int M, int K) {
    const int lane  = threadIdx.x & 31;
    const int wave  = threadIdx.x >> 5;
    const int mtile = blockIdx.x * 4 + wave;
    if (mtile * 16 >= M) return;                 // wave-uniform: EXEC stays full

    const int m0   = mtile * 16;
    const int hi   = lane >> 4;                  // 0 or 1
    const int lidx = lane & 15;                  // m (for A) / n (for B,C)

    const float* Arow = A + (size_t)(m0 + lidx) * K + 2 * hi;

    for (int ng = 0; ng < 4; ++ng) {
        const int nbase = ng * 64;
        v8f acc0 = {}, acc1 = {}, acc2 = {}, acc3 = {};

        for (int k0 = 0; k0 < K; k0 += 4) {
            v2f a = *(const v2f*)(Arow + k0);

            const float* Bp = B + (size_t)(k0 + 2 * hi) * OUT_DIM + nbase + lidx;
            v2f b0, b1, b2, b3;
            b0.x = Bp[0];   b0.y = Bp[OUT_DIM];
            b1.x = Bp[16];  b1.y = Bp[16 + OUT_DIM];
            b2.x = Bp[32];  b2.y = Bp[32 + OUT_DIM];
            b3.x = Bp[48];  b3.y = Bp[48 + OUT_DIM];

            acc0 = __builtin_amdgcn_wmma_f32_16x16x4_f32(false, a, false, b0,
                                                         (short)0, acc0, false, false);
            acc1 = __builtin_amdgcn_wmma_f32_16x16x4_f32(false, a, false, b1,
                                                         (short)0, acc1, false, false);
            acc2 = __builtin_amdgcn_wmma_f32_16x16x4_f32(false, a, false, b2,
                                                         (short)0, acc2, false, false);
            acc3 = __builtin_amdgcn_wmma_f32_16x16x4_f32(false, a, false, b3,
                                                         (short)0, acc3, false, false);
        }

        float* Crow = C + (size_t)(m0 + 8 * hi) * OUT_DIM + nbase + lidx;
#pragma unroll
        for (int v = 0; v < 8; ++v) {
            Crow[(size_t)v * OUT_DIM +  0] = acc0[v];
            Crow[(size_t)v * OUT_DIM + 16] = acc1[v];
            Crow[(size_t)v * OUT_DIM + 32] = acc2[v];
            Crow[(size_t)v * OUT_DIM + 48] = acc3[v];
        }
    }
}

// ---------------------------------------------------------------------------
// SpMM + ReLU (block-per-row). Edge (col,val) staged in LDS; each of the 256
// lanes owns one output column -> every edge gather is one coalesced 1KB row
// read served from L2 (51.2 MB source fits the 192 MB L2).
// When `att` is non-null, fuses the final attention-weighted combine.
// ---------------------------------------------------------------------------
__global__ void __launch_bounds__(256)
gcn_spmm_relu(const int* __restrict__ rowptr, const int* __restrict__ cols,
              const float* __restrict__ vals, const float* __restrict__ T,
              float* __restrict__ O,
              const float* __restrict__ o1, const float* __restrict__ o2,
              const float* __restrict__ att) {
    const int r = blockIdx.x;
    const int d = threadIdx.x;

    __shared__ int   s_col[256];
    __shared__ float s_val[256];

    const int e0 = rowptr[r];
    const int e1 = rowptr[r + 1];

    float acc = 0.0f;
    for (int base = e0; base < e1; base += 256) {
        const int cnt = min(256, e1 - base);
        __syncthreads();
        if (d < cnt) {
            s_col[d] = cols[base + d];
            s_val[d] = vals[base + d];
        }
        __syncthreads();
        for (int j = 0; j < cnt; ++j) {
            acc = fmaf(s_val[j], T[(size_t)s_col[j] * OUT_DIM + d], acc);
        }
    }

    const float r3  = fmaxf(acc, 0.0f);
    const size_t ix = (size_t)r * OUT_DIM + d;
    if (att) {
        O[ix] = att[0] * o1[ix] + att[1] * o2[ix] + att[2] * r3;
    } else {
        O[ix] = r3;
    }
}

// ---------------------------------------------------------------------------
// Launcher
// ---------------------------------------------------------------------------
extern "C" void kernel_launch(void* const* d_in, const int* in_sizes, int n_in,
                              void* d_out, int out_size, void* d_ws, size_t ws_size,
                              hipStream_t stream) {
    const float* x        = (const float*)d_in[0];
    const float* w1       = (const float*)d_in[1];
    const float* w2       = (const float*)d_in[2];
    const float* w3       = (const float*)d_in[3];
    const float* att      = (const float*)d_in[4];
    const float* adj_vals = (const float*)d_in[5];
    const int*   adj_rows = (const int*)d_in[6];
    const int*   adj_cols = (const int*)d_in[7];
    float*       out      = (float*)d_out;

    // Workspace layout: rowptr | t | o1 | o2   (~154 MB)
    const size_t SL = (size_t)N_NODES * OUT_DIM;          // elements per matrix
    char*  ws     = (char*)d_ws;
    int*   rowptr = (int*)ws;
    float* t      = (float*)(ws + (((size_t)(N_NODES + 1) * 4 + 255) & ~(size_t)255));
    float* o1     = t + SL;
    float* o2     = o1 + SL;

    // 1) CSR row pointers from sorted adj_rows
    gcn_build_rowptr<<<(N_NODES + 256) / 256, 256, 0, stream>>>(
        adj_rows, rowptr, N_NODES, N_EDGES);

    const int mtiles = (N_NODES + 15) / 16;               // 3125
    const int gblk   = (mtiles + 3) / 4;                  // 4 waves / block

    // 2) hop 1
    gcn_gemm_wmma_f32<<<gblk, 128, 0, stream>>>(x, w1, t, N_NODES, IN_DIM);
    gcn_spmm_relu<<<N_NODES, 256, 0, stream>>>(rowptr, adj_cols, adj_vals,
                                               t, o1, nullptr, nullptr, nullptr);
    // 3) hop 2
    gcn_gemm_wmma_f32<<<gblk, 128, 0, stream>>>(o1, w2, t, N_NODES, OUT_DIM);
    gcn_spmm_relu<<<N_NODES, 256, 0, stream>>>(rowptr, adj_cols, adj_vals,
                                               t, o2, nullptr, nullptr, nullptr);
    // 4) hop 3 + fused attention combine -> d_out
    gcn_gemm_wmma_f32<<<gblk, 128, 0, stream>>>(o2, w3, t, N_NODES, OUT_DIM);
    gcn_spmm_relu<<<N_NODES, 256, 0, stream>>>(rowptr, adj_cols, adj_vals,
                                               t, out, o1, o2, att);
}